// Interpolator_40896678592580
// MI455X (gfx1250) — compile-verified
//
#include <hip/hip_runtime.h>
#include <stdint.h>

// Problem constants from the reference (B,NT,N,NH,NV,F / NL,T,S).
constexpr int Bc  = 2;
constexpr int NTc = 2;
constexpr int Nc  = 16384;
constexpr int NHc = 4;
constexpr int NVc = 4;
constexpr int Fc  = 16;
constexpr int NLc = 256;
constexpr int Tc  = 64;
constexpr int Sc  = 256;         // N*NH/NL
constexpr int CH  = 128;         // source chunk staged in LDS per pass
constexpr int ROWP = CH + 1;     // padded LDS row stride (floats) -> bank-conflict free
constexpr size_t XI_ELEMS = (size_t)Bc * NTc * NLc * Tc * NVc * Fc;  // x_inter elems

typedef unsigned int u32x4 __attribute__((ext_vector_type(4)));
typedef int          i32x4 __attribute__((ext_vector_type(4)));
typedef int          i32x8 __attribute__((ext_vector_type(8)));

// Issue one TDM tile move: 64 rows x 128 floats (row stride 256 floats in global),
// with LDS padding of 1 DWORD after every 128 DWORDs (-> 129-float LDS rows).
__device__ __forceinline__ void tdm_load_tile(unsigned lds_byte, uint64_t gaddr_byte)
{
#if __has_builtin(__builtin_amdgcn_tensor_load_to_lds)
    u32x4 g0;
    g0.x = 1u;                                   // count=1 (valid descriptor)
    g0.y = lds_byte;                             // lds_addr
    g0.z = (unsigned)(gaddr_byte & 0xffffffffu); // global_addr[31:0]
    g0.w = (unsigned)((gaddr_byte >> 32) & 0x01ffffffu) | 0x80000000u; // [56:32] | type=2

    i32x8 g1;
    g1.s0 = 0x01920000;          // data_size=4B, pad_enable, pad_interval=128dw, pad=1dw
    g1.s1 = (int)(128u << 16);   // tensor_dim0[15:0]=128 (in bits 63:48)
    g1.s2 = (int)(64u  << 16);   // tensor_dim1[15:0]=64  (in bits 95:80)
    g1.s3 = (int)(128u << 16);   // tile_dim0=128         (bits 127:112)
    g1.s4 = 64;                  // tile_dim1=64           (bits 143:128)
    g1.s5 = 256;                 // tensor_dim0_stride=256 (bits 207:160)
    g1.s6 = 0;
    g1.s7 = 0;

    i32x4 z4 = {0, 0, 0, 0};
#if __clang_major__ >= 23
    i32x8 z8 = {0, 0, 0, 0, 0, 0, 0, 0};
    __builtin_amdgcn_tensor_load_to_lds(g0, g1, z4, z4, z8, 0);
#else
    __builtin_amdgcn_tensor_load_to_lds(g0, g1, z4, z4, 0);
#endif
#if __has_builtin(__builtin_amdgcn_s_wait_tensorcnt)
    __builtin_amdgcn_s_wait_tensorcnt(0);
#else
    asm volatile("s_wait_tensorcnt 0" ::: "memory");
#endif
#else
    // Fallback: wave0's 32 lanes copy the 32KB chunk with async b128 ops.
    const int lane = (int)(threadIdx.x & 31);
    for (int r = 0; r < Tc; ++r) {
        unsigned l = lds_byte + (unsigned)(r * ROWP * 4) + (unsigned)(lane * 16);
        unsigned gofs = (unsigned)(gaddr_byte & 0xffffffffu); // not used; fallback keeps asm form
        (void)gofs;
        asm volatile("global_load_async_to_lds_b128 %0, %1, %2 offset:0"
                     :: "v"(l),
                        "v"((unsigned)(r * Sc * 4 + lane * 16)),
                        "s"((const void*)(uintptr_t)(gaddr_byte))
                     : "memory");
    }
    asm volatile("s_wait_asynccnt 0" ::: "memory");
#endif
}

__global__ __launch_bounds__(256)
void interp_topk_kernel(const float* __restrict__ x,
                        const unsigned char* __restrict__ mask,
                        const float* __restrict__ dist,
                        const float* __restrict__ cutoff_p,
                        const int*   __restrict__ nnh_p,
                        const int*   __restrict__ power_p,
                        const float* __restrict__ mval_p,
                        float* __restrict__ out)
{
    __shared__ float        dist_lds[Tc * ROWP];   // 64 * 129 * 4B = 33,024 B
    __shared__ float        maskadd[Sc * NVc];     // 4,096 B
    __shared__ unsigned int maskraw[Sc];           // 1,024 B raw mask bytes

    const int tid = threadIdx.x;            // 0..255
    const int blk = blockIdx.x;             // 0..B*NT*NL-1
    const int l   = blk % NLc;
    const int g   = blk / NLc;              // b*NT + nt
    const int b   = g / NTc;

    const float cutoff = cutoff_p[0];
    const int   K      = nnh_p[0];          // reference uses 3; up to 4 supported
    const int   P      = power_p[0];
    const float mval   = mval_p[0];

    // ---- async-stage the block's 1KB mask slice into LDS (64 lanes x B128) ----
    if (tid < 64) {
        const unsigned lmask = (unsigned)(uintptr_t)maskraw + (unsigned)(tid * 16);
        const unsigned goff  =
            (unsigned)(((size_t)g * (Nc * NHc) + (size_t)l * Sc) * NVc) + (unsigned)(tid * 16);
        asm volatile("global_load_async_to_lds_b128 %0, %1, %2 offset:0"
                     :: "v"(lmask), "v"(goff), "s"(mask)
                     : "memory");
        asm volatile("s_wait_asynccnt 0" ::: "memory");
    }
    __syncthreads();

    // ---- convert mask bytes -> additive distance penalty ----
    {
        const int s = tid;
        const unsigned m = maskraw[s];
        maskadd[s * NVc + 0] = (m & 0x000000ffu) ? mval : 0.0f;
        maskadd[s * NVc + 1] = (m & 0x0000ff00u) ? mval : 0.0f;
        maskadd[s * NVc + 2] = (m & 0x00ff0000u) ? mval : 0.0f;
        maskadd[s * NVc + 3] = (m & 0xff000000u) ? mval : 0.0f;
    }

    const int t  = tid >> 2;                // target row 0..63
    const int nv = tid & 3;                 // variable 0..3

    const unsigned lds_base = (unsigned)(uintptr_t)dist_lds;
    const size_t tile_elem = ((size_t)(b * NLc + l) * Tc) * Sc;   // row0 of block tile

    // top-4 ascending (strict '<' => stable on ties like jax top_k)
    float d0 = 3.0e38f, d1 = 3.0e38f, d2 = 3.0e38f, d3 = 3.0e38f;
    int   i0 = 0, i1 = 0, i2 = 0, i3 = 0;

    for (int c = 0; c < Sc / CH; ++c) {
        __syncthreads();   // chunk buffer free (all readers of previous chunk done)

        if (tid < 32) {    // one wave drives the TDM; EXEC full, branch is wave-uniform
            const uint64_t gaddr =
                (uint64_t)(uintptr_t)dist + ((uint64_t)tile_elem + (uint64_t)c * CH) * 4u;
            tdm_load_tile(lds_base, gaddr);
        }
        __syncthreads();   // tile visible to all waves

        const float* __restrict__ drow = &dist_lds[t * ROWP];
        const int sbase = c * CH;
        for (int j = 0; j < CH; ++j) {
            const float d = drow[j] + maskadd[(sbase + j) * NVc + nv];
            const int   s = sbase + j;
            if (d < d3) {
                if (d < d2) {
                    d3 = d2; i3 = i2;
                    if (d < d1) {
                        d2 = d1; i2 = i1;
                        if (d < d0) { d1 = d0; i1 = i0; d0 = d; i0 = s; }
                        else        { d1 = d;  i1 = s; }
                    } else { d2 = d; i2 = s; }
                } else { d3 = d; i3 = s; }
            }
        }
    }

    // ---- weights: clamp, inverse power, normalize over first K neighbors ----
    float dd[4] = { d0, d1, d2, d3 };
    int   ii[4] = { i0, i1, i2, i3 };
    float w[4];
    float wsum = 0.0f;
    #pragma unroll
    for (int k = 0; k < 4; ++k) {
        if (k < K) {
            float dk = fmaxf(dd[k], cutoff);
            dd[k] = dk;
            float wp = 1.0f;
            for (int p = 0; p < P; ++p) wp *= dk;
            const float wk = 1.0f / wp;
            w[k] = wk;
            wsum += wk;
        } else {
            w[k] = 0.0f;
        }
    }
    const float inv = 1.0f / wsum;

    // ---- gather neighbor features (64B-aligned float4 loads) and accumulate ----
    float4 a0 = {0,0,0,0}, a1 = {0,0,0,0}, a2 = {0,0,0,0}, a3 = {0,0,0,0};
    #pragma unroll
    for (int k = 0; k < 4; ++k) {
        if (k < K) {
            const float wk = w[k] * inv;
            const float4* xp = reinterpret_cast<const float4*>(
                x + (((size_t)g * (Nc * NHc) + (size_t)l * Sc + ii[k]) * NVc + nv) * Fc);
            const float4 v0 = xp[0], v1 = xp[1], v2 = xp[2], v3 = xp[3];
            a0.x += wk * v0.x; a0.y += wk * v0.y; a0.z += wk * v0.z; a0.w += wk * v0.w;
            a1.x += wk * v1.x; a1.y += wk * v1.y; a1.z += wk * v1.z; a1.w += wk * v1.w;
            a2.x += wk * v2.x; a2.y += wk * v2.y; a2.z += wk * v2.z; a2.w += wk * v2.w;
            a3.x += wk * v3.x; a3.y += wk * v3.y; a3.z += wk * v3.z; a3.w += wk * v3.w;
        }
    }

    // ---- outputs ----
    const size_t lt = (size_t)l * Tc + t;
    float4* o1 = reinterpret_cast<float4*>(
        out + (((size_t)g * (NLc * Tc) + lt) * NVc + nv) * Fc);
    o1[0] = a0; o1[1] = a1; o1[2] = a2; o1[3] = a3;

    float* o2 = out + XI_ELEMS + (((size_t)g * (NLc * Tc) + lt) * (size_t)K) * NVc + nv;
    #pragma unroll
    for (int k = 0; k < 4; ++k) {
        if (k < K) o2[(size_t)k * NVc] = dd[k];
    }
}

extern "C" void kernel_launch(void* const* d_in, const int* in_sizes, int n_in,
                              void* d_out, int out_size, void* d_ws, size_t ws_size,
                              hipStream_t stream) {
    (void)in_sizes; (void)n_in; (void)out_size; (void)d_ws; (void)ws_size;
    const float*         x      = (const float*)d_in[0];
    const unsigned char* mask   = (const unsigned char*)d_in[1];   // jax bool -> byte
    const float*         dist   = (const float*)d_in[2];
    const float*         cutoff = (const float*)d_in[3];
    const int*           nnh    = (const int*)d_in[4];
    const int*           power  = (const int*)d_in[5];
    const float*         mval   = (const float*)d_in[6];

    dim3 grid(Bc * NTc * NLc);   // 1024 blocks: one (b, nt, l) group each
    dim3 block(256);             // 8 wave32 waves; thread = (t, nv)
    hipLaunchKernelGGL(interp_topk_kernel, grid, block, 0, stream,
                       x, mask, dist, cutoff, nnh, power, mval, (float*)d_out);
}